// ComputeNodeAreaFromRouteMap_6356551598556
// MI455X (gfx1250) — compile-verified
//
#include <hip/hip_runtime.h>

// ---- Problem constants (match reference) ----
#define NBX 512
#define NBY 512
#define KW  4
static constexpr float XL_F  = 0.0f;
static constexpr float YL_F  = 0.0f;
static constexpr float BSX_F = 1000.0f / 512.0f;   // 1.953125 = 125/64, exact in fp32
static constexpr float BSY_F = 1000.0f / 512.0f;
static constexpr float INV_BSX_F = 512.0f / 1000.0f;  // 0.512f (approx reciprocal)
static constexpr float INV_BSY_F = 512.0f / 1000.0f;

// Exact floor(v / BS) for v >= 0 without the IEEE divide sequence:
// approximate with reciprocal multiply, then fix up against the exact bin
// boundaries. (float)i * BS is exact for i <= 511 since BS = 125/64.
__device__ __forceinline__ int exact_bin_floor(float v, float bs, float inv_bs) {
    int i = (int)(v * inv_bs);           // v >= 0 -> trunc == floor (approx)
    float bl = (float)i * bs;            // exact product
    i += (bl + bs <= v) ? 1 : 0;         // approx was one too low
    i -= (bl > v) ? 1 : 0;               // approx was one too high
    return i;
}

// One node per lane. Memory/gather-bound kernel: streamed inputs use NT hints
// so the 1MB utilization map stays resident in WGP$/L2; map gathers use
// default (RT) temporal policy; rows are prefetched (global_prefetch_b8)
// after the overlap arithmetic, just before the gathers.
__global__ __launch_bounds__(256) void node_area_from_route_map_kernel(
    const float* __restrict__ pos,
    const float* __restrict__ node_size_x,
    const float* __restrict__ node_size_y,
    const float* __restrict__ umap,   // [NBX][NBY] row-major
    float* __restrict__ out,
    int n, int half)
{
    int i = blockIdx.x * blockDim.x + threadIdx.x;
    if (i >= n) return;

    // Streamed, single-use data -> non-temporal loads (TH=NT).
    float x  = __builtin_nontemporal_load(&pos[i]);
    float y  = __builtin_nontemporal_load(&pos[half + i]);
    float sx = __builtin_nontemporal_load(&node_size_x[i]);
    float sy = __builtin_nontemporal_load(&node_size_y[i]);

    float xh = x + sx;
    float yh = y + sy;

    // bin0 = clip(floor((lo - origin)/bin_size), 0, NB-1)  (reference order)
    int bx0 = exact_bin_floor(x - XL_F, BSX_F, INV_BSX_F);
    bx0 = bx0 < 0 ? 0 : (bx0 > NBX - 1 ? NBX - 1 : bx0);
    int by0 = exact_bin_floor(y - YL_F, BSY_F, INV_BSY_F);
    by0 = by0 < 0 ? 0 : (by0 > NBY - 1 ? NBY - 1 : by0);

    // Per-axis: unclamped idx for validity, clamped idx for the gather,
    // overlap zeroed when idx is out of range (matches reference exactly).
    float ovx[KW], ovy[KW];
    int   ixc[KW], iyc[KW];

#pragma unroll
    for (int k = 0; k < KW; ++k) {
        int idx  = bx0 + k;               // >= 0 by construction
        float bl = XL_F + (float)idx * BSX_F;
        float bh = bl + BSX_F;
        float ov = fminf(xh, bh) - fmaxf(x, bl);
        ovx[k] = (idx < NBX) ? fmaxf(ov, 0.0f) : 0.0f;
        ixc[k] = idx > NBX - 1 ? NBX - 1 : idx;
    }
#pragma unroll
    for (int k = 0; k < KW; ++k) {
        int idx  = by0 + k;
        float bl = YL_F + (float)idx * BSY_F;
        float bh = bl + BSY_F;
        float ov = fminf(yh, bh) - fmaxf(y, bl);
        ovy[k] = (idx < NBY) ? fmaxf(ov, 0.0f) : 0.0f;
        iyc[k] = idx > NBY - 1 ? NBY - 1 : idx;
    }

    // Prefetch the four map rows we are about to gather from (the 4 y-values
    // per row span <= 16B from iyc[0], usually one cacheline).
    // Lowers to global_prefetch_b8 on gfx1250.
#pragma unroll
    for (int a = 0; a < KW; ++a) {
        __builtin_prefetch(&umap[ixc[a] * NBY + iyc[0]], 0, 3);
    }

    // acc = sum_a ovx[a] * ( sum_b U[ix[a], iy[b]] * ovy[b] )
    float acc = 0.0f;
#pragma unroll
    for (int a = 0; a < KW; ++a) {
        const float* __restrict__ row = umap + ixc[a] * NBY;
        float sa = 0.0f;
#pragma unroll
        for (int b = 0; b < KW; ++b) {
            sa = fmaf(row[iyc[b]], ovy[b], sa);
        }
        acc = fmaf(sa, ovx[a], acc);
    }

    // Streaming output -> non-temporal store.
    __builtin_nontemporal_store(acc, &out[i]);
}

extern "C" void kernel_launch(void* const* d_in, const int* in_sizes, int n_in,
                              void* d_out, int out_size, void* d_ws, size_t ws_size,
                              hipStream_t stream) {
    const float* pos  = (const float*)d_in[0];   // 2*NUM_MOVABLE
    const float* nsx  = (const float*)d_in[1];   // NUM_MOVABLE
    const float* nsy  = (const float*)d_in[2];   // NUM_MOVABLE
    const float* umap = (const float*)d_in[3];   // NBX*NBY
    float* out = (float*)d_out;                  // NUM_MOVABLE

    int n    = in_sizes[1];       // NUM_MOVABLE
    int half = in_sizes[0] / 2;   // offset of y within pos

    const int threads = 256;
    int blocks = (n + threads - 1) / threads;
    node_area_from_route_map_kernel<<<blocks, threads, 0, stream>>>(
        pos, nsx, nsy, umap, out, n, half);
}